// MonaUncertaintyPrompt_67843303408332
// MI455X (gfx1250) — compile-verified
//
#include <hip/hip_runtime.h>
#include <hip/hip_bf16.h>
#include <math.h>

typedef float v2f __attribute__((ext_vector_type(2)));
typedef float v8f __attribute__((ext_vector_type(8)));
typedef unsigned int u32;
typedef u32 u32x4 __attribute__((ext_vector_type(4)));
typedef int i32x4 __attribute__((ext_vector_type(4)));
typedef int i32x8 __attribute__((ext_vector_type(8)));

static constexpr int B_ = 8, C_ = 512, Hh = 64, Ww = 64;
static constexpr int N_ = Hh * Ww;        // 4096
static constexpr int BN = B_ * N_;        // 32768
static constexpr int Cd = 64, Kp = 100, KP = 112, P_ = 64;
static constexpr int C3 = 192, C2 = 128;

__device__ __forceinline__ v8f wmma4(v2f a, v2f b, v8f c) {
  // V_WMMA_F32_16X16X4_F32 : D(16x16,f32) = A(16x4,f32) x B(4x16,f32) + C
  // lane fragment (wave32): half = lane>>4, lq = lane&15, k0 = 2*half
  //   a = {A[lq][k0], A[lq][k0+1]}   b = {B[k0][lq], B[k0+1][lq]}
  //   d[j] = D[8*half + j][lq]
  return __builtin_amdgcn_wmma_f32_16x16x4_f32(false, a, false, b, (short)0, c,
                                               false, false);
}

// Tensor Data Mover: 2D tile load Global -> LDS (D# per CDNA5 ISA).
// data_size=4B. Optional LDS padding: pad_amount_code+1 dwords inserted after
// every (2<<pad_interval_code) dwords. 6-arg builtin form (this toolchain).
__device__ __forceinline__ void tdm_load_2d(u32 lds_off, const void* gptr,
                                            u32 tile_d0, u32 tile_d1,
                                            u32 tensor_d0, u32 tensor_d1,
                                            u32 stride_d0, u32 pad_iv_code,
                                            u32 pad_amt_code, bool pad_en) {
  unsigned long long ga = (unsigned long long)gptr;
  u32x4 g0;
  g0[0] = 1u;                                     // count=1, user desc, no gather
  g0[1] = lds_off;                                // lds_addr (bytes)
  g0[2] = (u32)(ga & 0xffffffffu);                // global_addr[31:0]
  g0[3] = (u32)((ga >> 32) & 0x01ffffffu) |       // global_addr[56:32]
          (2u << 30);                             // type=2 ("image")
  i32x8 g1;
  g1[0] = (int)((2u << 16) |                      // data_size = 4B
                ((pad_en ? 1u : 0u) << 20) |      // pad_enable
                (pad_iv_code << 22) |             // pad_interval
                (pad_amt_code << 25));            // pad_amount
  g1[1] = (int)((tensor_d0 & 0xffffu) << 16);                       // dim0[15:0]
  g1[2] = (int)(((tensor_d0 >> 16) & 0xffffu) |
                ((tensor_d1 & 0xffffu) << 16));   // dim0[31:16] | dim1[15:0]
  g1[3] = (int)(((tensor_d1 >> 16) & 0xffffu) |
                ((tile_d0 & 0xffffu) << 16));     // dim1[31:16] | tile_dim0
  g1[4] = (int)(tile_d1 & 0xffffu);               // tile_dim1, tile_dim2=0
  g1[5] = (int)stride_d0;                         // tensor_dim0_stride[31:0]
  g1[6] = 0;
  g1[7] = 0;
  i32x4 gz4 = {0, 0, 0, 0};                       // groups 2/3 unused (2D)
  i32x8 gz8 = {0, 0, 0, 0, 0, 0, 0, 0};
  __builtin_amdgcn_tensor_load_to_lds(g0, g1, gz4, gz4, gz8, 0);
}

// ---------------------------------------------------------------- kernel 1
// depthwise 3x3 conv (SAME) + bias + SiLU ; ir (B,192,64,64) -> dwout same
__global__ void k_dw(const float* __restrict__ ir, const float* __restrict__ w,
                     const float* __restrict__ bias, float* __restrict__ outp) {
  int idx = blockIdx.x * 256 + threadIdx.x;         // B*C3*N exactly
  int hw = idx & (N_ - 1);
  int bc = idx >> 12;
  int c = bc % C3;
  int b = bc / C3;
  int h = hw >> 6, wq = hw & 63;
  const float* src = ir + ((size_t)b * C3 + c) * N_;
  const float* w9 = w + c * 9;
  float acc = bias[c];
#pragma unroll
  for (int dh = -1; dh <= 1; ++dh) {
    int hh = h + dh;
    if (hh < 0 || hh >= Hh) continue;
#pragma unroll
    for (int dw = -1; dw <= 1; ++dw) {
      int wwp = wq + dw;
      if (wwp < 0 || wwp >= Ww) continue;
      acc += w9[(dh + 1) * 3 + (dw + 1)] * src[hh * Ww + wwp];
    }
  }
  outp[idx] = acc / (1.0f + __expf(-acc));          // SiLU
}

// ---------------------------------------------------------------- kernel 2
// 1x1 conv 192->128 as WMMA GEMM. One wave = 16(out ch) x 16(spatial) tile.
// gb layout: (BN, 128), first 64 = gamma, last 64 = beta.
__global__ void k_pw(const float* __restrict__ dwout,
                     const float* __restrict__ pw_w,
                     const float* __restrict__ pw_b, float* __restrict__ gb) {
  int wv = threadIdx.x >> 5;
  int lane = threadIdx.x & 31;
  int half = lane >> 4, lq = lane & 15, k0 = 2 * half;
  int wg = blockIdx.x * 8 + wv;                      // 0..16383
  int mTile = wg & 7, sTile = wg >> 3;
  int m0 = mTile * 16;
  int r0 = sTile * 16;                               // BN row; never crosses batch
  int b = r0 >> 12, n0 = r0 & (N_ - 1);
  const float* Bbase = dwout + (size_t)b * C3 * N_ + n0 + lq;
  const float* Abase = pw_w + (size_t)(m0 + lq) * C3;
  v8f acc = {};
  for (int k = 0; k < C3; k += 4) {
    v2f a = *(const v2f*)(Abase + k + k0);
    v2f bb;
    bb.x = Bbase[(size_t)(k + k0) * N_];
    bb.y = Bbase[(size_t)(k + k0 + 1) * N_];
    acc = wmma4(a, bb, acc);
  }
#pragma unroll
  for (int j = 0; j < 8; ++j) {
    int m = m0 + 8 * half + j;                       // output channel
    gb[(size_t)(r0 + lq) * C2 + m] = acc[j] + pw_b[m];
  }
}

// ---------------------------------------------------------------- kernel 3
// prompt_k / prompt_v (100x64, zero-padded to 112) + 1/||prompt_k||
__global__ void k_prompt(const float* __restrict__ prompts,
                         const float* __restrict__ pk_w,
                         const float* __restrict__ pk_b,
                         const float* __restrict__ pv_w,
                         const float* __restrict__ pv_b, float* __restrict__ pk,
                         float* __restrict__ pv, float* __restrict__ pkinv) {
  int tid = threadIdx.x;
  for (int idx = tid; idx < KP * Cd; idx += 256) {
    int kk = idx / Cd, cd = idx % Cd;
    float ak = 0.f, av = 0.f;
    if (kk < Kp) {
      const float* pr = prompts + kk * P_;
      const float* wk = pk_w + cd * P_;
      const float* wv = pv_w + cd * P_;
      for (int p = 0; p < P_; ++p) { ak += pr[p] * wk[p]; av += pr[p] * wv[p]; }
      ak += pk_b[cd]; av += pv_b[cd];
    }
    pk[idx] = ak; pv[idx] = av;
  }
  __syncthreads();
  if (tid < KP) {
    float s = 0.f;
    for (int cd = 0; cd < Cd; ++cd) { float v = pk[tid * Cd + cd]; s += v * v; }
    pkinv[tid] = (tid < Kp) ? 1.0f / fmaxf(sqrtf(s), 1e-12f) : 0.0f;
  }
}

// ---------------------------------------------------------------- kernel 4
// per 16-row tile: TDM-load x tile -> LN(512) -> WMMA 512->64 + FiLM
//                 -> WMMA q 64->64 -> 1/||q||
// LDS x tile layout (written by TDM with pad 1dw/16dw): lx[c*17 + i]
__global__ void __launch_bounds__(128)
k_lntd(const float* __restrict__ x, const float* __restrict__ ln_w,
       const float* __restrict__ ln_b, const float* __restrict__ td_w,
       const float* __restrict__ td_b, const float* __restrict__ q_w,
       const float* __restrict__ q_b, const float* __restrict__ gb,
       float* __restrict__ xt, float* __restrict__ qbuf,
       float* __restrict__ qinv) {
  __shared__ float lx[C_ * 17];                      // 34816 B
  __shared__ float lxt[16 * 65];
  __shared__ float lqs[16 * 65];
  __shared__ float rsum[16][8], rsq[16][8];
  __shared__ float mu_s[16], rstd_s[16];
  int tid = threadIdx.x;                             // 128 threads = 4 waves
  int r0 = blockIdx.x * 16;
  int b = r0 >> 12, n0 = r0 & (N_ - 1);
  const float* xb = x + (size_t)b * C_ * N_ + n0;    // NCHW tile start
  int wv = tid >> 5, lane = tid & 31, half = lane >> 4, lq = lane & 15;
  int k0 = 2 * half;
  if (wv == 0) {
    // DMA: 16(spatial, contiguous) x 512(channel, stride 4096) f32 tile,
    // pad 1 dword after every 16 -> LDS row stride 17 floats.
    tdm_load_2d((u32)(size_t)lx, xb, /*tile*/16, C_,
                /*tensor*/(u32)N_, C_, /*stride0*/(u32)N_,
                /*pad: 16 dw*/3, /*1 dw*/0, true);
    __builtin_amdgcn_s_wait_tensorcnt((short)0);
  }
  __syncthreads();
  {
    int i = tid >> 3, s = tid & 7;                   // 8 partials per row i
    float sm = 0.f, sq = 0.f;
    for (int c = s * 64; c < s * 64 + 64; ++c) {
      float v = lx[c * 17 + i];
      sm += v; sq += v * v;
    }
    rsum[i][s] = sm; rsq[i][s] = sq;
  }
  __syncthreads();
  if (tid < 16) {
    float sm = 0.f, sq = 0.f;
    for (int s = 0; s < 8; ++s) { sm += rsum[tid][s]; sq += rsq[tid][s]; }
    float mu = sm / C_;
    mu_s[tid] = mu;
    rstd_s[tid] = rsqrtf(sq / C_ - mu * mu + 1e-6f);
  }
  __syncthreads();
  for (int idx = tid; idx < 16 * C_; idx += 128) {
    int i = idx & 15, c = idx >> 4;                  // 16 lanes -> contiguous LDS
    lx[c * 17 + i] = (lx[c * 17 + i] - mu_s[i]) * rstd_s[i] * ln_w[c] + ln_b[c];
  }
  __syncthreads();
  {  // td linear 512->64, wave wv owns output cols wv*16..+15 ; then FiLM
    int ncol = wv * 16 + lq;
    const float* wrow = td_w + (size_t)ncol * C_;
    v8f acc = {};
    for (int k = 0; k < C_; k += 4) {
      // A[m=i][k=c] = lx[c*17 + i]; lanes 0-15 -> contiguous, conflict-free
      v2f a; a.x = lx[(k + k0) * 17 + lq]; a.y = lx[(k + k0 + 1) * 17 + lq];
      v2f bb = *(const v2f*)(wrow + k + k0);
      acc = wmma4(a, bb, acc);
    }
    float biasv = td_b[ncol];
#pragma unroll
    for (int j = 0; j < 8; ++j) {
      int m = 8 * half + j;
      int r = r0 + m;
      float g = gb[(size_t)r * C2 + ncol];
      float bt = gb[(size_t)r * C2 + 64 + ncol];
      float v = (acc[j] + biasv) * (1.0f + g) + bt;
      lxt[m * 65 + ncol] = v;
      xt[(size_t)r * Cd + ncol] = v;
    }
  }
  __syncthreads();
  {  // q = x_trans @ q_w^T + q_b
    int ncol = wv * 16 + lq;
    const float* wrow = q_w + (size_t)ncol * Cd;
    v8f acc = {};
    for (int k = 0; k < Cd; k += 4) {
      v2f a; a.x = lxt[lq * 65 + k + k0]; a.y = lxt[lq * 65 + k + k0 + 1];
      v2f bb = *(const v2f*)(wrow + k + k0);
      acc = wmma4(a, bb, acc);
    }
    float biasv = q_b[ncol];
#pragma unroll
    for (int j = 0; j < 8; ++j) {
      int m = 8 * half + j;
      float v = acc[j] + biasv;
      lqs[m * 65 + ncol] = v;
      qbuf[(size_t)(r0 + m) * Cd + ncol] = v;
    }
  }
  __syncthreads();
  if (tid < 16) {
    float s = 0.f;
    for (int c = 0; c < Cd; ++c) { float v = lqs[tid * 65 + c]; s += v * v; }
    qinv[r0 + tid] = 1.0f / fmaxf(sqrtf(s), 1e-12f);
  }
}

// ---------------------------------------------------------------- kernel 5
// TDM-load q tile -> cos-sim (WMMA) -> uncertainty + masked softmax
// -> attn@prompt_v (WMMA) -> blend
__global__ void __launch_bounds__(128)
k_attn(const float* __restrict__ qbuf, const float* __restrict__ qinv,
       const float* __restrict__ pk, const float* __restrict__ pkinv,
       const float* __restrict__ pv, const float* __restrict__ xt,
       const float* __restrict__ temperature, float* __restrict__ xc,
       float* __restrict__ uout) {
  __shared__ float lq[16 * 65];                      // TDM: pad 1dw/64dw
  __shared__ float lcos[16 * 113];
  __shared__ float qi[16], den[16], uu[16];
  int tid = threadIdx.x;                             // 128
  int r0 = blockIdx.x * 16;
  int wv = tid >> 5, lane = tid & 31, half = lane >> 4, lq16 = lane & 15;
  int k0 = 2 * half;
  if (wv == 0) {
    // q rows r0..r0+15 are one contiguous 4KB region: 1-row tile of 1024 dw,
    // pad 1 dword after every 64 -> LDS layout lq[i*65 + c].
    tdm_load_2d((u32)(size_t)lq, qbuf + (size_t)r0 * Cd, /*tile*/16 * Cd, 1,
                /*tensor*/16 * Cd, 1, /*stride0*/16 * Cd,
                /*pad: 64 dw*/5, /*1 dw*/0, true);
    __builtin_amdgcn_s_wait_tensorcnt((short)0);
  }
  if (tid < 16) qi[tid] = qinv[r0 + tid];
  __syncthreads();
  for (int t = wv; t < 7; t += 4) {                  // wave-uniform loop
    int ncol = t * 16 + lq16;
    const float* pkrow = pk + (size_t)ncol * Cd;
    v8f acc = {};
    for (int k = 0; k < Cd; k += 4) {
      v2f a; a.x = lq[lq16 * 65 + k + k0]; a.y = lq[lq16 * 65 + k + k0 + 1];
      v2f bb = *(const v2f*)(pkrow + k + k0);
      acc = wmma4(a, bb, acc);
    }
    float ps = pkinv[ncol];
#pragma unroll
    for (int j = 0; j < 8; ++j) {
      int m = 8 * half + j;
      lcos[m * 113 + ncol] = acc[j] * qi[m] * ps;
    }
  }
  __syncthreads();
  if (tid < 16) {
    float mx = -1e30f;
    for (int k = 0; k < Kp; ++k) mx = fmaxf(mx, lcos[tid * 113 + k]);
    float u = 1.0f - (mx + 1.0f) * 0.5f;
    uu[tid] = u;
    uout[r0 + tid] = u;
    float invT = 1.0f / temperature[0];
    float s = 0.f;
    for (int k = 0; k < Kp; ++k) {
      float e = __expf((lcos[tid * 113 + k] - mx) * invT);
      lcos[tid * 113 + k] = e;
      s += e;
    }
    for (int k = Kp; k < KP; ++k) lcos[tid * 113 + k] = 0.f;  // mask padding
    den[tid] = s;
  }
  __syncthreads();
  {  // x_proto = attn @ prompt_v, then blend with x_trans
    int ncol = wv * 16 + lq16;
    v8f acc = {};
    for (int k = 0; k < KP; k += 4) {
      v2f a; a.x = lcos[lq16 * 113 + k + k0]; a.y = lcos[lq16 * 113 + k + k0 + 1];
      v2f bb;
      bb.x = pv[(size_t)(k + k0) * Cd + ncol];
      bb.y = pv[(size_t)(k + k0 + 1) * Cd + ncol];
      acc = wmma4(a, bb, acc);
    }
#pragma unroll
    for (int j = 0; j < 8; ++j) {
      int m = 8 * half + j;
      int r = r0 + m;
      float xp = acc[j] / den[m];
      float u = uu[m];
      float xv = xt[(size_t)r * Cd + ncol];
      xc[(size_t)r * Cd + ncol] = (1.0f - u) * xv + u * xp;
    }
  }
}

// ---------------------------------------------------------------- kernel 6
// out = x_cal @ proj_w^T + proj_b + x   (NCHW). Wave = 16(spatial)x16(chan).
__global__ void k_proj(const float* __restrict__ xc,
                       const float* __restrict__ proj_w,
                       const float* __restrict__ proj_b,
                       const float* __restrict__ x, float* __restrict__ out) {
  int wv = threadIdx.x >> 5, lane = threadIdx.x & 31;
  int half = lane >> 4, lq = lane & 15, k0 = 2 * half;
  int wg = blockIdx.x * 8 + wv;                      // 0..65535
  int sTile = wg >> 5, cTile = wg & 31;
  int r0 = sTile * 16, c0 = cTile * 16;
  const float* arow = xc + (size_t)(r0 + lq) * Cd;
  const float* brow = proj_w + (size_t)(c0 + lq) * Cd;
  v8f acc = {};
  for (int k = 0; k < Cd; k += 4) {
    v2f a = *(const v2f*)(arow + k + k0);
    v2f bb = *(const v2f*)(brow + k + k0);
    acc = wmma4(a, bb, acc);
  }
  int c = c0 + lq;
  float biasv = proj_b[c];
  int b = r0 >> 12, n0 = r0 & (N_ - 1);
  const float* xsrc = x + (size_t)b * C_ * N_ + (size_t)c * N_ + n0;
  float* dst = out + (size_t)b * C_ * N_ + (size_t)c * N_ + n0;
#pragma unroll
  for (int j = 0; j < 8; ++j) {
    int m = 8 * half + j;                            // contiguous in n
    dst[m] = acc[j] + biasv + xsrc[m];
  }
}

// ---------------------------------------------------------------- launcher
extern "C" void kernel_launch(void* const* d_in, const int* in_sizes, int n_in,
                              void* d_out, int out_size, void* d_ws,
                              size_t ws_size, hipStream_t stream) {
  const float* x       = (const float*)d_in[0];
  const float* ir      = (const float*)d_in[1];
  const float* dw_w    = (const float*)d_in[2];
  const float* dw_b    = (const float*)d_in[3];
  const float* pw_w    = (const float*)d_in[4];
  const float* pw_b    = (const float*)d_in[5];
  const float* ln_w    = (const float*)d_in[6];
  const float* ln_b    = (const float*)d_in[7];
  const float* td_w    = (const float*)d_in[8];
  const float* td_b    = (const float*)d_in[9];
  const float* q_w     = (const float*)d_in[10];
  const float* q_b     = (const float*)d_in[11];
  const float* pk_w    = (const float*)d_in[12];
  const float* pk_b    = (const float*)d_in[13];
  const float* pv_w    = (const float*)d_in[14];
  const float* pv_b    = (const float*)d_in[15];
  const float* prompts = (const float*)d_in[16];
  const float* proj_w  = (const float*)d_in[17];
  const float* proj_b  = (const float*)d_in[18];
  const float* temp    = (const float*)d_in[19];
  float* out = (float*)d_out;
  float* uout = out + (size_t)B_ * C_ * N_;          // second tuple output

  char* wsp = (char*)d_ws;
  auto alloc = [&](size_t bytes) {
    float* p = (float*)wsp;
    wsp += (bytes + 255) & ~(size_t)255;
    return p;
  };
  float* dwout = alloc((size_t)B_ * C3 * N_ * 4);    // 25.2 MB
  float* gb    = alloc((size_t)BN * C2 * 4);         // 16.8 MB
  float* xt    = alloc((size_t)BN * Cd * 4);         //  8.4 MB
  float* qbuf  = alloc((size_t)BN * Cd * 4);         //  8.4 MB
  float* qiv   = alloc((size_t)BN * 4);
  float* pk    = alloc((size_t)KP * Cd * 4);
  float* pv    = alloc((size_t)KP * Cd * 4);
  float* pkinv = alloc((size_t)KP * 4);
  float* xcb   = alloc((size_t)BN * Cd * 4);         //  8.4 MB

  k_dw<<<(B_ * C3 * N_) / 256, 256, 0, stream>>>(ir, dw_w, dw_b, dwout);
  k_pw<<<(8 * (BN / 16)) / 8, 256, 0, stream>>>(dwout, pw_w, pw_b, gb);
  k_prompt<<<1, 256, 0, stream>>>(prompts, pk_w, pk_b, pv_w, pv_b, pk, pv, pkinv);
  k_lntd<<<BN / 16, 128, 0, stream>>>(x, ln_w, ln_b, td_w, td_b, q_w, q_b, gb,
                                      xt, qbuf, qiv);
  k_attn<<<BN / 16, 128, 0, stream>>>(qbuf, qiv, pk, pkinv, pv, xt, temp, xcb,
                                      uout);
  k_proj<<<((BN / 16) * 32) / 8, 256, 0, stream>>>(xcb, proj_w, proj_b, x, out);
}